// QuantumClassifierModel_65481071403600
// MI455X (gfx1250) — compile-verified
//
#include <hip/hip_runtime.h>
#include <hip/hip_bf16.h>

typedef __attribute__((ext_vector_type(2))) float v2f;
typedef __attribute__((ext_vector_type(8))) float v8f;

namespace {
constexpr int NQ = 20;     // qubits
constexpr int BATCH = 32;  // batch size
}

// One wave32 does everything: closed-form Pauli-propagated features, then
// logits = feat(32x20) @ W^T(20x2) + b via V_WMMA_F32_16X16X4_F32 (fp32 exact).
__global__ __launch_bounds__(32) void
qcls_pauli_wmma_kernel(const float* __restrict__ x,      // (32, 20)
                       const float* __restrict__ theta,  // (2, 20)
                       const float* __restrict__ W,      // (2, 20)
                       const float* __restrict__ bias,   // (2,)
                       float* __restrict__ out) {        // (32, 2)
  __shared__ float s_c0[NQ], s_s0[NQ], s_c1[NQ], s_s1[NQ];
  __shared__ float s_w[2][NQ];            // s_w[n][q] = W[n][q]
  __shared__ float s_feat[BATCH][NQ];     // features per batch row

  const int lane = threadIdx.x;           // 0..31, exactly one wave32

  // --- per-qubit trig of the ansatz angles (shared across batch) ---
  if (lane < NQ) {
    const float t0 = theta[lane];         // theta[0][q]
    const float t1 = theta[NQ + lane];    // theta[1][q]
    s_c0[lane] = cosf(t0);
    s_s0[lane] = sinf(t0);
    s_c1[lane] = cosf(t1);
    s_s1[lane] = sinf(t1);
    s_w[0][lane] = W[lane];
    s_w[1][lane] = W[NQ + lane];
  }
  __syncthreads();

  // --- closed-form features: lane = batch index ---
  // <Z_q> = c1q*c0q*ux_q - s1q*(c0_{q-1}ux_{q-1})*(s0q*ux_q)*(c0_{q+1}ux_{q+1})
  // feature = -<Z_q>   (boundary neighbor factors = 1)
  float u[NQ], g[NQ];
#pragma unroll
  for (int q = 0; q < NQ; ++q) u[q] = cosf(x[lane * NQ + q]);
#pragma unroll
  for (int q = 0; q < NQ; ++q) g[q] = s_c0[q] * u[q];
#pragma unroll
  for (int q = 0; q < NQ; ++q) {
    const float gL = (q > 0) ? g[q - 1] : 1.0f;
    const float gR = (q < NQ - 1) ? g[q + 1] : 1.0f;
    s_feat[lane][q] = s_s1[q] * gL * (s_s0[q] * u[q]) * gR - s_c1[q] * g[q];
  }
  __syncthreads();

  // --- logits via fp32 WMMA 16x16x4, two M-tiles (rows 0-15, 16-31), K=20=5*4 ---
  const int half = lane >> 4;   // 0: lanes 0-15, 1: lanes 16-31
  const int nidx = lane & 15;   // N column for B/C/D, M row index for A
  v8f acc0 = {};
  v8f acc1 = {};
#pragma unroll
  for (int k0 = 0; k0 < NQ; k0 += 4) {
    const int kr = k0 + 2 * half;  // this lane's K rows: kr, kr+1
    v2f a0, a1, bb;
    // A layout (32-bit 16x4): lanes 0-15 -> K=k0,k0+1; lanes 16-31 -> K=k0+2,k0+3
    a0.x = s_feat[nidx][kr];
    a0.y = s_feat[nidx][kr + 1];
    a1.x = s_feat[16 + nidx][kr];
    a1.y = s_feat[16 + nidx][kr + 1];
    // B layout (4x16): row K striped across lanes (N = nidx); only N<2 nonzero
    const int nn = (nidx < 2) ? nidx : 0;
    const float w0 = s_w[nn][kr];
    const float w1 = s_w[nn][kr + 1];
    bb.x = (nidx < 2) ? w0 : 0.0f;
    bb.y = (nidx < 2) ? w1 : 0.0f;
    acc0 = __builtin_amdgcn_wmma_f32_16x16x4_f32(false, a0, false, bb,
                                                 (short)0, acc0, false, false);
    acc1 = __builtin_amdgcn_wmma_f32_16x16x4_f32(false, a1, false, bb,
                                                 (short)0, acc1, false, false);
  }

  // --- D layout: element v holds row M = 8*half + v, column N = nidx ---
  if (nidx < 2) {
    const float bn = bias[nidx];
#pragma unroll
    for (int v = 0; v < 8; ++v) {
      const int m = 8 * half + v;
      out[m * 2 + nidx] = acc0[v] + bn;          // batch rows 0..15
      out[(16 + m) * 2 + nidx] = acc1[v] + bn;   // batch rows 16..31
    }
  }
}

extern "C" void kernel_launch(void* const* d_in, const int* in_sizes, int n_in,
                              void* d_out, int out_size, void* d_ws, size_t ws_size,
                              hipStream_t stream) {
  (void)in_sizes; (void)n_in; (void)out_size; (void)d_ws; (void)ws_size;
  const float* x     = (const float*)d_in[0];  // (32, 20)
  const float* theta = (const float*)d_in[1];  // (2, 20)
  const float* W     = (const float*)d_in[2];  // (2, 20)
  const float* b     = (const float*)d_in[3];  // (2,)
  float* out = (float*)d_out;                  // (32, 2)
  qcls_pauli_wmma_kernel<<<1, 32, 0, stream>>>(x, theta, W, b, out);
}